// GCN_5471788335195
// MI455X (gfx1250) — compile-verified
//
#include <hip/hip_runtime.h>
#include <hip/hip_bf16.h>
#include <math.h>

typedef __attribute__((ext_vector_type(2))) float v2f;
typedef __attribute__((ext_vector_type(8))) float v8f;

#define FDIM   128
#define NCLASS 40

// ---------------- degree / norm ----------------
__global__ void k_zero(float* __restrict__ p, int n) {
    int i = blockIdx.x * blockDim.x + threadIdx.x;
    if (i < n) p[i] = 0.0f;
}

__global__ void k_count(const int* __restrict__ dst, float* __restrict__ deg, int E) {
    int e = blockIdx.x * blockDim.x + threadIdx.x;
    if (e < E) atomicAdd(&deg[dst[e]], 1.0f);
}

__global__ void k_dinv(float* __restrict__ d, int n) {
    int i = blockIdx.x * blockDim.x + threadIdx.x;
    if (i < n) d[i] = rsqrtf(d[i] + 1.0f);   // +1 self loop, deg >= 1
}

// ---------------- fp32 WMMA GEMM: H[nrows x 128] = X[nrows x 128] @ W[128 x 128] ----------------
// One block = 64 output rows (4 row-tiles). 8 waves; wave w owns column tile w.
// Each wave loads its 32 B fragments (64 VGPRs) once and reuses them across 4 row-tiles.
// A tile staged in LDS via float4 (global_load_b128), stride 132 -> conflict-free ds reads.
__global__ __launch_bounds__(256) void k_gemm_x128(const float* __restrict__ X,
                                                   const float* __restrict__ W,
                                                   float* __restrict__ H, int nrows) {
    __shared__ float lds[64 * 132];
    const int base = blockIdx.x * 64;

    // cooperative load of 64x128 A tile, float4 granularity (rows clamped for the tail block)
    for (int i = threadIdx.x; i < 64 * 32; i += 256) {
        int r = i >> 5, c4 = i & 31;
        int rg = base + r;
        if (rg >= nrows) rg = nrows - 1;
        const float4 v = *(const float4*)(X + (size_t)rg * FDIM + c4 * 4);
        float* p = &lds[r * 132 + c4 * 4];
        p[0] = v.x; p[1] = v.y; p[2] = v.z; p[3] = v.w;
    }
    __syncthreads();

    const int lane  = threadIdx.x & 31;
    const int wave  = threadIdx.x >> 5;      // column tile 0..7
    const int m     = lane & 15;             // A row within tile / B col within tile
    const int khalf = (lane >> 4) * 2;       // K sub-pair select: 0 or 2
    const int col   = wave * 16 + m;
    const int rhalf = (lane >> 4) * 8;       // D row half select

#if __has_builtin(__builtin_amdgcn_wmma_f32_16x16x4_f32)
    // B fragments register-cached: reused by all 4 row-tiles
    v2f bfrag[32];
    #pragma unroll
    for (int kk = 0; kk < 32; ++kk) {
        const int k0 = kk * 4 + khalf;
        bfrag[kk].x = W[(size_t)k0 * FDIM + col];
        bfrag[kk].y = W[(size_t)(k0 + 1) * FDIM + col];
    }
    for (int rt = 0; rt < 4; ++rt) {
        const int rbase = rt * 16;
        v8f acc = {};
        #pragma unroll
        for (int kk = 0; kk < 32; ++kk) {
            const int k0 = kk * 4 + khalf;
            v2f a;
            a.x = lds[(rbase + m) * 132 + k0];
            a.y = lds[(rbase + m) * 132 + k0 + 1];
            // (neg_a, A, neg_b, B, c_mod, C, reuse_a, reuse_b)
            acc = __builtin_amdgcn_wmma_f32_16x16x4_f32(false, a, false, bfrag[kk],
                                                        (short)0, acc, false, false);
        }
        #pragma unroll
        for (int v = 0; v < 8; ++v) {
            int row = base + rbase + rhalf + v;
            if (row < nrows) H[(size_t)row * FDIM + col] = acc[v];
        }
    }
#else
    for (int rt = 0; rt < 4; ++rt) {
        const int rbase = rt * 16;
        #pragma unroll
        for (int v = 0; v < 8; ++v) {
            float s = 0.0f;
            for (int k = 0; k < FDIM; ++k)
                s += lds[(rbase + rhalf + v) * 132 + k] * W[(size_t)k * FDIM + col];
            int row = base + rbase + rhalf + v;
            if (row < nrows) H[(size_t)row * FDIM + col] = s;
        }
    }
#endif
}

// ---------------- self-loop init: agg = h * dinv^2 ----------------
__global__ void k_selfloop(const float* __restrict__ h, const float* __restrict__ dinv,
                           float* __restrict__ agg, int total) {
    int i = blockIdx.x * blockDim.x + threadIdx.x;
    if (i < total) {
        int node = i >> 7;
        float di = dinv[node];
        agg[i] = h[i] * di * di;
    }
}

// ---------------- edge scatter: one wave32 per 4 edges, coalesced 128B atomic bursts ----------------
__global__ void k_scatter(const float* __restrict__ h, const float* __restrict__ dinv,
                          const int* __restrict__ src, const int* __restrict__ dst,
                          float* __restrict__ agg, int E) {
    int tid  = blockIdx.x * blockDim.x + threadIdx.x;
    int lane = tid & 31;
    int e0   = (tid >> 5) * 4;
    #pragma unroll
    for (int j = 0; j < 4; ++j) {
        int e = e0 + j;
        if (e >= E) return;
        int s = src[e], d = dst[e];
        // prefetch next edge's source row while this one is in flight
        if (j < 3 && e + 1 < E) {
            int sn = src[e + 1];
            __builtin_prefetch(h + (size_t)sn * FDIM + lane * 4, 0, 0);
        }
        float coef = dinv[s] * dinv[d];
        const float* hs = h + (size_t)s * FDIM;
        float* ad = agg + (size_t)d * FDIM;
        #pragma unroll
        for (int i = 0; i < 4; ++i) {
            int f = lane + 32 * i;
            atomicAdd(&ad[f], hs[f] * coef);
        }
    }
}

// ---------------- bias + relu in place ----------------
__global__ void k_bias_relu(float* __restrict__ a, const float* __restrict__ b, int total) {
    int i = blockIdx.x * blockDim.x + threadIdx.x;
    if (i < total) a[i] = fmaxf(a[i] + b[i & 127], 0.0f);
}

// ---------------- head: logits = [h1 h2] @ Wlin + blin, then log_softmax ----------------
// one wave32 per node; lane owns class `lane` and (lane<8) class `lane+32`
__global__ __launch_bounds__(256) void k_head(const float* __restrict__ h1,
                                              const float* __restrict__ h2,
                                              const float* __restrict__ Wlin,
                                              const float* __restrict__ blin,
                                              float* __restrict__ out, int n) {
    int lane = threadIdx.x & 31;
    int node = blockIdx.x * 8 + (threadIdx.x >> 5);
    if (node >= n) return;

    const float* r1 = h1 + (size_t)node * FDIM;
    const float* r2 = h2 + (size_t)node * FDIM;
    const int c0 = lane, c1 = lane + 32;
    const bool has1 = (c1 < NCLASS);

    float l0 = blin[c0];
    float l1 = has1 ? blin[c1] : 0.0f;
    for (int k = 0; k < FDIM; ++k) {
        float xv = r1[k];
        const float* wr = Wlin + (size_t)k * NCLASS;
        l0 += xv * wr[c0];
        if (has1) l1 += xv * wr[c1];
    }
    for (int k = 0; k < FDIM; ++k) {
        float xv = r2[k];
        const float* wr = Wlin + (size_t)(k + FDIM) * NCLASS;
        l0 += xv * wr[c0];
        if (has1) l1 += xv * wr[c1];
    }

    float mx = has1 ? fmaxf(l0, l1) : l0;
    for (int o = 16; o > 0; o >>= 1) mx = fmaxf(mx, __shfl_xor(mx, o, 32));
    float s = __expf(l0 - mx) + (has1 ? __expf(l1 - mx) : 0.0f);
    for (int o = 16; o > 0; o >>= 1) s += __shfl_xor(s, o, 32);
    float ls = __logf(s);

    out[(size_t)node * NCLASS + c0] = l0 - mx - ls;
    if (has1) out[(size_t)node * NCLASS + c1] = l1 - mx - ls;
}

extern "C" void kernel_launch(void* const* d_in, const int* in_sizes, int n_in,
                              void* d_out, int out_size, void* d_ws, size_t ws_size,
                              hipStream_t stream) {
    const float* x    = (const float*)d_in[0];
    const int*   ei   = (const int*)  d_in[1];
    const float* W1   = (const float*)d_in[2];
    const float* b1   = (const float*)d_in[3];
    const float* W2   = (const float*)d_in[4];
    const float* b2   = (const float*)d_in[5];
    const float* Wlin = (const float*)d_in[6];
    const float* blin = (const float*)d_in[7];
    float* out = (float*)d_out;

    const int n = in_sizes[0] / FDIM;      // 100000
    const int E = in_sizes[1] / 2;         // 1600000
    const int* src = ei;
    const int* dst = ei + E;

    float* ws   = (float*)d_ws;
    float* dinv = ws;                                  // n floats (deg -> dinv in place)
    float* htmp = dinv + n;                            // n*128
    float* h1   = htmp + (size_t)n * FDIM;             // n*128
    float* h2   = h1   + (size_t)n * FDIM;             // n*128

    const int total = n * FDIM;
    const dim3 blk(256);
    const int gN   = (n + 255) / 256;
    const int gE   = (E + 255) / 256;
    const int gTot = (total + 255) / 256;
    const int gSc  = (((E + 3) / 4) * 32 + 255) / 256; // one wave per 4 edges
    const int gMM  = (n + 63) / 64;                    // 64 rows per block

    // normalization
    k_zero<<<gN, blk, 0, stream>>>(dinv, n);
    k_count<<<gE, blk, 0, stream>>>(dst, dinv, E);
    k_dinv<<<gN, blk, 0, stream>>>(dinv, n);

    // layer 1
    k_gemm_x128<<<gMM, blk, 0, stream>>>(x, W1, htmp, n);
    k_selfloop<<<gTot, blk, 0, stream>>>(htmp, dinv, h1, total);
    k_scatter<<<gSc, blk, 0, stream>>>(htmp, dinv, src, dst, h1, E);
    k_bias_relu<<<gTot, blk, 0, stream>>>(h1, b1, total);

    // layer 2
    k_gemm_x128<<<gMM, blk, 0, stream>>>(h1, W2, htmp, n);
    k_selfloop<<<gTot, blk, 0, stream>>>(htmp, dinv, h2, total);
    k_scatter<<<gSc, blk, 0, stream>>>(htmp, dinv, src, dst, h2, E);
    k_bias_relu<<<gTot, blk, 0, stream>>>(h2, b2, total);

    // head + log_softmax
    k_head<<<(n + 7) / 8, blk, 0, stream>>>(h1, h2, Wlin, blin, out, n);
}